// BodyModel_88313117540378
// MI455X (gfx1250) — compile-verified
//
#include <hip/hip_runtime.h>
#include <math.h>

typedef __attribute__((ext_vector_type(2))) float v2f;
typedef __attribute__((ext_vector_type(8))) float v8f;
typedef __attribute__((ext_vector_type(2))) int   v2i;

#ifndef __has_builtin
#define __has_builtin(x) 0
#endif
#if __has_builtin(__builtin_amdgcn_global_load_async_to_lds_b64) && \
    __has_builtin(__builtin_amdgcn_s_wait_asynccnt)
#define USE_ASYNC 1
#else
#define USE_ASYNC 0
#endif

#define BATCH 512
#define NV    6890
#define NJ    24
#define NBETA 10
#define NP    207
#define PF_PAD 208
#define NCOL  (NV * 3)   // 20670

// d_out float offsets (verts, posed_joints, A, transforms, v_posed)
#define OFF_VERTS 0
#define OFF_PJ    (BATCH * NV * 3)
#define OFF_A     (OFF_PJ + BATCH * NJ * 3)
#define OFF_TF    (OFF_A + BATCH * NJ * 16)
#define OFF_VP    (OFF_TF + BATCH * NJ * 16)

// workspace float offsets
#define WS_JTJS 0
#define WS_PF   800      // BATCH * PF_PAD

#if USE_ASYNC
#define WAIT_ASYNC(n) __builtin_amdgcn_s_wait_asynccnt(n)
#else
#define WAIT_ASYNC(n)
#endif

// async (or fallback) 8-byte global -> LDS copy
__device__ __forceinline__ void cp_b64(const float* g, float* l) {
#if USE_ASYNC
  __builtin_amdgcn_global_load_async_to_lds_b64(
      (__attribute__((address_space(1))) v2i*)(g),
      (__attribute__((address_space(3))) v2i*)(l), 0, 0);
#else
  *(v2f*)l = *(const v2f*)g;
#endif
}

// ---------------------------------------------------------------------------
// K0: JT[j][k] (cols 0..2) and JS[j][k][l] (cols 3+k*10+l) = Jreg-reductions
// ---------------------------------------------------------------------------
__global__ void k0_jreg(const float* __restrict__ vt,
                        const float* __restrict__ sd,
                        const float* __restrict__ jreg,
                        float* __restrict__ jtjs) {
  __shared__ float red[256 * 33];
  const int j = blockIdx.x;
  const int t = threadIdx.x;
  float acc[33];
#pragma unroll
  for (int c = 0; c < 33; ++c) acc[c] = 0.f;
  for (int v = t; v < NV; v += 256) {
    const float w = jreg[j * NV + v];
#pragma unroll
    for (int k = 0; k < 3; ++k) acc[k] += w * vt[v * 3 + k];
#pragma unroll
    for (int k = 0; k < 3; ++k)
#pragma unroll
      for (int l = 0; l < 10; ++l)
        acc[3 + k * 10 + l] += w * sd[v * 30 + k * 10 + l];
  }
  for (int c = 0; c < 33; ++c) red[t * 33 + c] = acc[c];
  __syncthreads();
  for (int s = 128; s > 0; s >>= 1) {
    if (t < s)
      for (int c = 0; c < 33; ++c) red[t * 33 + c] += red[(t + s) * 33 + c];
    __syncthreads();
  }
  if (t < 33) jtjs[j * 33 + t] = red[t];
}

// ---------------------------------------------------------------------------
// K1: per-batch rodrigues + pose_feature + Jloc + kinematic chain (wave32/b)
// ---------------------------------------------------------------------------
__global__ void k1_chain(const float* __restrict__ betas,
                         const float* __restrict__ pose,
                         const int* __restrict__ parents,
                         const float* __restrict__ jtjs,
                         float* __restrict__ pf,
                         float* __restrict__ out) {
  __shared__ float tm[NJ * 16];
  __shared__ float T [NJ * 16];
  __shared__ float jl[NJ * 3];
  const int b = blockIdx.x;
  const int lane = threadIdx.x;

  if (lane < NJ) {
    const int j = lane;
    const float ax = pose[b * 72 + j * 3 + 0];
    const float ay = pose[b * 72 + j * 3 + 1];
    const float az = pose[b * 72 + j * 3 + 2];
    const float ex = ax + 1e-8f, ey = ay + 1e-8f, ez = az + 1e-8f;
    const float ang = sqrtf(ex * ex + ey * ey + ez * ez);
    const float rx = ax / ang, ry = ay / ang, rz = az / ang;
    const float c = cosf(ang), s = sinf(ang), o = 1.f - c;
    const float R00 = 1.f - o * (ry * ry + rz * rz);
    const float R01 = -s * rz + o * rx * ry;
    const float R02 =  s * ry + o * rx * rz;
    const float R10 =  s * rz + o * rx * ry;
    const float R11 = 1.f - o * (rx * rx + rz * rz);
    const float R12 = -s * rx + o * ry * rz;
    const float R20 = -s * ry + o * rx * rz;
    const float R21 =  s * rx + o * ry * rz;
    const float R22 = 1.f - o * (rx * rx + ry * ry);

    if (j >= 1) {
      float* p = pf + b * PF_PAD + (j - 1) * 9;
      p[0] = R00 - 1.f; p[1] = R01;       p[2] = R02;
      p[3] = R10;       p[4] = R11 - 1.f; p[5] = R12;
      p[6] = R20;       p[7] = R21;       p[8] = R22 - 1.f;
    } else {
      pf[b * PF_PAD + 207] = 0.f;   // zero K-pad column
    }

#pragma unroll
    for (int k = 0; k < 3; ++k) {
      float v = jtjs[j * 33 + k];
#pragma unroll
      for (int l = 0; l < NBETA; ++l)
        v += betas[b * NBETA + l] * jtjs[j * 33 + 3 + k * 10 + l];
      jl[j * 3 + k] = v;
    }
    tm[j * 16 + 0] = R00; tm[j * 16 + 1] = R01; tm[j * 16 + 2]  = R02;
    tm[j * 16 + 4] = R10; tm[j * 16 + 5] = R11; tm[j * 16 + 6]  = R12;
    tm[j * 16 + 8] = R20; tm[j * 16 + 9] = R21; tm[j * 16 + 10] = R22;
    tm[j * 16 + 12] = 0.f; tm[j * 16 + 13] = 0.f;
    tm[j * 16 + 14] = 0.f; tm[j * 16 + 15] = 1.f;
  }
  __syncthreads();
  if (lane < NJ) {
    const int j = lane;
    float tx = jl[j * 3 + 0], ty = jl[j * 3 + 1], tz = jl[j * 3 + 2];
    if (j > 0) {
      const int p = parents[j];
      tx -= jl[p * 3 + 0]; ty -= jl[p * 3 + 1]; tz -= jl[p * 3 + 2];
    }
    tm[j * 16 + 3] = tx; tm[j * 16 + 7] = ty; tm[j * 16 + 11] = tz;
  }
  __syncthreads();
  if (lane < 16) T[lane] = tm[lane];
  __syncthreads();
  for (int i = 1; i < NJ; ++i) {
    const int par = parents[i];
    if (lane < 16) {
      const int p = lane >> 2, q = lane & 3;
      float acc = 0.f;
#pragma unroll
      for (int m = 0; m < 4; ++m)
        acc += T[par * 16 + p * 4 + m] * tm[i * 16 + m * 4 + q];
      T[i * 16 + lane] = acc;
    }
    __syncthreads();
  }

  float* outTF = out + OFF_TF + (size_t)b * NJ * 16;
  float* outA  = out + OFF_A  + (size_t)b * NJ * 16;
  float* outPJ = out + OFF_PJ + (size_t)b * NJ * 3;
  for (int idx = lane; idx < NJ * 16; idx += 32) outTF[idx] = T[idx];
  if (lane < NJ) {
    const int j = lane;
    outPJ[j * 3 + 0] = T[j * 16 + 3];
    outPJ[j * 3 + 1] = T[j * 16 + 7];
    outPJ[j * 3 + 2] = T[j * 16 + 11];
    const float jx = jl[j * 3 + 0], jy = jl[j * 3 + 1], jz = jl[j * 3 + 2];
#pragma unroll
    for (int p = 0; p < 4; ++p) {
      const float ib = T[j * 16 + p * 4 + 0] * jx +
                       T[j * 16 + p * 4 + 1] * jy +
                       T[j * 16 + p * 4 + 2] * jz;
#pragma unroll
      for (int q = 0; q < 4; ++q)
        outA[j * 16 + p * 4 + q] = T[j * 16 + p * 4 + q] - ((q == 3) ? ib : 0.f);
    }
  }
}

// ---------------------------------------------------------------------------
// posedirs K-chunk staging: 52 rows x 48 cols into an LDS buffer, zero-padded.
// Async engine writes and DS-pipe zero stores hit disjoint addresses.
// ---------------------------------------------------------------------------
__device__ __forceinline__ void stage_pd_chunk(const float* __restrict__ pd,
                                               float* __restrict__ buf,
                                               int krow0, int n0, int lane,
                                               bool tail) {
  const int rows = (NP - krow0 < 52) ? (NP - krow0) : 52;
  if (tail) {
#pragma unroll 1
    for (int i = lane; i < 52 * 24; i += 32) {
      const int r = i / 24, cp = i - r * 24;
      const bool written = (r < rows) && (n0 + cp * 2 + 1 < NCOL);
      if (!written) { buf[r * 48 + cp * 2] = 0.f; buf[r * 48 + cp * 2 + 1] = 0.f; }
    }
  } else if (rows < 52) {
    if (lane < 24) { buf[51 * 48 + 2 * lane] = 0.f; buf[51 * 48 + 2 * lane + 1] = 0.f; }
  }
  const int elems = rows * 24;   // b64 elements
#pragma unroll 1
  for (int i = 0; i < 39; ++i) {   // ceil(52*24/32): constant trip -> 39 async ops
    const int idx = i * 32 + lane;
    if (idx < elems) {
      const int r = idx / 24, cp = idx - r * 24;
      const int col = n0 + cp * 2;
      if (!tail || (col + 1) < NCOL)
        cp_b64(pd + (size_t)(krow0 + r) * NCOL + col, buf + r * 48 + cp * 2);
    }
  }
}

// ---------------------------------------------------------------------------
// K2: fused pose-GEMM (WMMA f32 16x16x4, LDS-resident, async double-buffered)
//     + shape blend + LBS epilogue. One wave32 per block; tile 16b x 48cols.
// ---------------------------------------------------------------------------
__global__ void k2_lbs(const float* __restrict__ betas,
                       const float* __restrict__ vt,
                       const float* __restrict__ sd,
                       const float* __restrict__ pd,      // posedirs (207,20670)
                       const float* __restrict__ lbsw,    // (V,24)
                       const float* __restrict__ pf,      // ws (B,208)
                       float* __restrict__ out) {
  __shared__ float smem[8320];   // 33.3 KB total
  float* pf_s = smem;            // [0,3328)  16x208
  float* pd0  = smem + 3328;     // [3328,5824)  52x48
  float* pd1  = smem + 5824;     // [5824,8320)  52x48
  // epilogue aliases (GEMM buffers dead by then):
  float* vp_s    = smem;         // 16x48
  float* betas_s = smem + 768;   // 16x10
  float* w_s     = smem + 928;   // 16x24
  float* A_s     = smem + 1312;  // 16x24x16

  const int lane = threadIdx.x;
  const int m0 = blockIdx.x * 16;
  const int n0 = blockIdx.y * 48;
  const int v0 = n0 / 3;
  const bool tail = (n0 + 48 > NCOL);

  // ---- stage pose_feature tile (contiguous) + first two pd chunks ----
#pragma unroll 1
  for (int i = lane; i < 16 * PF_PAD / 2; i += 32)   // 52 async ops
    cp_b64(pf + (size_t)m0 * PF_PAD + 2 * i, pf_s + 2 * i);
  stage_pd_chunk(pd, pd0, 0, n0, lane, tail);    // 39 async ops
  stage_pd_chunk(pd, pd1, 52, n0, lane, tail);   // 39 async ops

  // ---- GEMM: D = pose_feature(16x208) x posedirs(208x48), 4 chunks ----
  const int half_ = lane >> 4;
  const int lrow  = lane & 15;
  const float* pfrow = pf_s + lrow * PF_PAD;
  v8f c0 = {}, c1 = {}, c2 = {};

  for (int chunk = 0; chunk < 4; ++chunk) {
    if (chunk == 3) { WAIT_ASYNC(0); } else { WAIT_ASYNC(39); }
    __syncthreads();
    const float* buf = (chunk & 1) ? pd1 : pd0;
#pragma unroll
    for (int ks = 0; ks < 13; ++ks) {
      const int ka = chunk * 52 + ks * 4 + 2 * half_;
      const int lk = ks * 4 + 2 * half_;
      v2f a;
      a.x = pfrow[ka];
      a.y = pfrow[ka + 1];
      v2f b0, b1, b2;
      b0.x = buf[lk * 48 + lrow];      b0.y = buf[(lk + 1) * 48 + lrow];
      b1.x = buf[lk * 48 + 16 + lrow]; b1.y = buf[(lk + 1) * 48 + 16 + lrow];
      b2.x = buf[lk * 48 + 32 + lrow]; b2.y = buf[(lk + 1) * 48 + 32 + lrow];
      c0 = __builtin_amdgcn_wmma_f32_16x16x4_f32(false, a, false, b0, (short)0, c0, false, false);
      c1 = __builtin_amdgcn_wmma_f32_16x16x4_f32(false, a, false, b1, (short)0, c1, false, false);
      c2 = __builtin_amdgcn_wmma_f32_16x16x4_f32(false, a, false, b2, (short)0, c2, false, false);
    }
    if (chunk < 2) {
      __syncthreads();   // chunk's DS reads retired before buffer reuse
      stage_pd_chunk(pd, (chunk & 1) ? pd1 : pd0, (chunk + 2) * 52, n0, lane, tail);
    }
  }
  __syncthreads();   // GEMM LDS fully retired; safe to alias

  // ---- stage epilogue operands into aliased LDS ----
  const float* gA = out + OFF_A + (size_t)m0 * NJ * 16;
#pragma unroll 1
  for (int i = lane; i < 80; i += 32)
    cp_b64(betas + (size_t)m0 * NBETA + 2 * i, betas_s + 2 * i);
#pragma unroll 1
  for (int i = lane; i < 192; i += 32) {
    const int g = v0 * NJ + 2 * i;
    if (g + 1 < NV * NJ) cp_b64(lbsw + g, w_s + 2 * i);
    else { w_s[2 * i] = 0.f; w_s[2 * i + 1] = 0.f; }
  }
#pragma unroll 1
  for (int i = lane; i < 3072; i += 32)
    cp_b64(gA + 2 * i, A_s + 2 * i);
  WAIT_ASYNC(0);
  __syncthreads();

  // ---- epilogue 1: shape blend, emit v_posed, stash tile in LDS ----
  float* outVP = out + OFF_VP;
  v8f accv[3] = {c0, c1, c2};
#pragma unroll
  for (int sub = 0; sub < 3; ++sub) {
    const int n = n0 + sub * 16 + lrow;
    const bool nok = n < NCOL;
    float sdv[NBETA];
    float vtv = 0.f;
    if (nok) {
      vtv = vt[n];
#pragma unroll
      for (int l = 0; l < NBETA; ++l) sdv[l] = sd[(size_t)n * NBETA + l];
    } else {
#pragma unroll
      for (int l = 0; l < NBETA; ++l) sdv[l] = 0.f;
    }
#pragma unroll
    for (int r = 0; r < 8; ++r) {
      const int rowl = half_ * 8 + r;
      float vs = vtv;
#pragma unroll
      for (int l = 0; l < NBETA; ++l) vs += betas_s[rowl * NBETA + l] * sdv[l];
      const float vp = accv[sub][r] + vs;
      vp_s[rowl * 48 + sub * 16 + lrow] = vp;
      if (nok) outVP[(size_t)(m0 + rowl) * NCOL + n] = vp;
    }
  }
  __syncthreads();

  // ---- epilogue 2: LBS blend of A + vertex transform -> verts ----
  float* outV = out + OFF_VERTS;
  for (int it = 0; it < 8; ++it) {
    const int pr = it * 32 + lane;
    const int rowl = pr >> 4;
    const int vl = pr & 15;
    const int v = v0 + vl;
    if (v < NV) {
      const float vp0 = vp_s[rowl * 48 + vl * 3 + 0];
      const float vp1 = vp_s[rowl * 48 + vl * 3 + 1];
      const float vp2 = vp_s[rowl * 48 + vl * 3 + 2];
      float t[12];
#pragma unroll
      for (int e = 0; e < 12; ++e) t[e] = 0.f;
      for (int j = 0; j < NJ; ++j) {
        const float w = w_s[vl * NJ + j];
        const float* Aj = &A_s[rowl * (NJ * 16) + j * 16];
#pragma unroll
        for (int e = 0; e < 12; ++e) t[e] += w * Aj[e];
      }
      const size_t base = (size_t)(m0 + rowl) * NCOL + (size_t)v * 3;
      outV[base + 0] = t[0] * vp0 + t[1] * vp1 + t[2]  * vp2 + t[3];
      outV[base + 1] = t[4] * vp0 + t[5] * vp1 + t[6]  * vp2 + t[7];
      outV[base + 2] = t[8] * vp0 + t[9] * vp1 + t[10] * vp2 + t[11];
    }
  }
}

// ---------------------------------------------------------------------------
extern "C" void kernel_launch(void* const* d_in, const int* in_sizes, int n_in,
                              void* d_out, int out_size, void* d_ws, size_t ws_size,
                              hipStream_t stream) {
  const float* betas   = (const float*)d_in[0];
  const float* pose    = (const float*)d_in[1];
  const float* vtempl  = (const float*)d_in[2];
  const float* shaped  = (const float*)d_in[3];
  const float* posed   = (const float*)d_in[4];
  const float* jreg    = (const float*)d_in[5];
  const float* lbsw    = (const float*)d_in[6];
  const int*   parents = (const int*)d_in[7];
  float* out = (float*)d_out;
  float* ws  = (float*)d_ws;
  float* jtjs = ws + WS_JTJS;
  float* pf   = ws + WS_PF;

  k0_jreg<<<NJ, 256, 0, stream>>>(vtempl, shaped, jreg, jtjs);
  k1_chain<<<BATCH, 32, 0, stream>>>(betas, pose, parents, jtjs, pf, out);
  dim3 grid2(BATCH / 16, (NCOL + 47) / 48);
  k2_lbs<<<grid2, 32, 0, stream>>>(betas, vtempl, shaped, posed, lbsw, pf, out);
}